// _ES_21423296872652
// MI455X (gfx1250) — compile-verified
//
#include <hip/hip_runtime.h>
#include <cstdint>

#ifndef __has_builtin
#define __has_builtin(x) 0
#endif

#define ES_BATCH   16384
#define ES_NTIME   1024
#define ES_SEAS    7
#define ES_SLEN    (ES_NTIME + ES_SEAS)   /* 1031 */
#define ES_RPB     64                     /* rows per block (= threads per block) */
#define ES_TT      32                     /* time-tile width */
#define ES_PAD     33                     /* padded LDS row stride (dwords), coprime w/ 64 banks */
#define ES_NTILES  (ES_NTIME / ES_TT)     /* 32 */
#define ES_TILEF   (ES_RPB * ES_PAD)      /* floats per (padded) LDS tile = 2112 */

#if defined(__HIP_DEVICE_COMPILE__) && \
    __has_builtin(__builtin_amdgcn_global_load_async_to_lds_b32) && \
    __has_builtin(__builtin_amdgcn_global_store_async_from_lds_b32)
#define ES_ASYNC 1
#else
#define ES_ASYNC 0
#endif

#if ES_ASYNC && __has_builtin(__builtin_amdgcn_tensor_load_to_lds) && \
    __has_builtin(__builtin_amdgcn_s_wait_tensorcnt)
#define ES_TDM 1
#else
#define ES_TDM 0
#endif

typedef __attribute__((address_space(1))) int* es_gptr_t;
typedef __attribute__((address_space(3))) int* es_lptr_t;
typedef unsigned int es_u32x4 __attribute__((ext_vector_type(4)));
typedef int          es_i32x8 __attribute__((ext_vector_type(8)));
typedef int          es_i32x4 __attribute__((ext_vector_type(4)));

// Per-lane DMA: copies one dword from global address g to LDS address l  (ASYNCcnt)
__device__ __forceinline__ void es_async_load(const float* g, float* l) {
#if ES_ASYNC
  __builtin_amdgcn_global_load_async_to_lds_b32(
      (es_gptr_t)(uintptr_t)g, (es_lptr_t)(uint32_t)(uintptr_t)l, 0, 0);
#else
  *l = *g;
#endif
}

// Per-lane DMA: copies one dword from LDS address l to global address g  (ASYNCcnt)
__device__ __forceinline__ void es_async_store(float* g, const float* l) {
#if ES_ASYNC
  __builtin_amdgcn_global_store_async_from_lds_b32(
      (es_gptr_t)(uintptr_t)g, (es_lptr_t)(uint32_t)(uintptr_t)l, 0, 0);
#else
  *g = *l;
#endif
}

__device__ __forceinline__ void es_wait_async0() {
#if ES_ASYNC
#if __has_builtin(__builtin_amdgcn_s_wait_asynccnt)
  __builtin_amdgcn_s_wait_asynccnt(0);
#else
  asm volatile("s_wait_asynccnt 0x0" ::: "memory");
#endif
#endif
}

// TDM: DMA one 32-row x 32-col f32 tile (row stride 1024 elems) into LDS with
// 1-dword padding every 32 dwords -> LDS row stride 33 (bank-conflict free).
__device__ __forceinline__ void es_tdm_load_tile(const float* gbase, const float* lds_dst) {
#if ES_TDM
  const uint64_t ga = (uint64_t)(uintptr_t)gbase;
  const uint32_t la = (uint32_t)(uintptr_t)lds_dst;
  es_u32x4 g0;
  g0[0] = 1u;                                    // count=1, user descriptor
  g0[1] = la;                                    // lds_addr (bytes)
  g0[2] = (uint32_t)ga;                          // global_addr[31:0]
  g0[3] = (uint32_t)(ga >> 32) | 0x80000000u;    // global_addr[56:32] | type=2
  es_i32x8 g1;
  g1[0] = (2 << 16)        // data_size = 4B
        | (1 << 20)        // pad_enable
        | (4 << 22)        // pad_interval: 32 dwords
        | (0 << 25);       // pad_amount: 1 dword
  g1[1] = (int)((unsigned)ES_NTIME << 16);       // tensor_dim0[15:0]=1024
  g1[2] = (int)((unsigned)ES_BATCH << 16);       // tensor_dim0 hi=0 | tensor_dim1 lo=16384
  g1[3] = (int)((unsigned)ES_TT << 16);          // tensor_dim1 hi=0 | tile_dim0=32
  g1[4] = 32;                                    // tile_dim1=32 rows, tile_dim2=0
  g1[5] = ES_NTIME;                              // tensor_dim0_stride lo = 1024
  g1[6] = 0;                                     // stride hi / dim1_stride lo
  g1[7] = 0;
  es_i32x4 g2; g2[0] = 1; g2[1] = 1; g2[2] = 0; g2[3] = 0;  // dim2=1, dim3=1, tile_dim3=0
  es_i32x4 g3; g3[0] = 0; g3[1] = (1 << 16); g3[2] = 0; g3[3] = 0;  // dim4=1, tile_dim4=0
  es_i32x8 g4; g4[0] = 0; g4[1] = 0; g4[2] = 0; g4[3] = 0;
               g4[4] = 0; g4[5] = 0; g4[6] = 0; g4[7] = 0;  // trailing group unused
  __builtin_amdgcn_tensor_load_to_lds(g0, g1, g2, g3, g4, 0);
#else
  (void)gbase; (void)lds_dst;
#endif
}

// Issue the y-tile fetch for time-offset t0 into ybuf.
__device__ __forceinline__ void es_issue_tile_load(const float* y, int row0, int w, int l,
                                                   int t0, float* ybuf) {
#if ES_TDM
  // one TDM op per wave, each wave loads its own 32 rows
  es_tdm_load_tile(y + (size_t)(row0 + w * 32) * ES_NTIME + t0,
                   ybuf + (size_t)(w * 32) * ES_PAD);
#else
  for (int p = 0; p < ES_RPB / 2; ++p) {
    int rp = p * 2 + w;
    es_async_load(y + (size_t)(row0 + rp) * ES_NTIME + t0 + l, ybuf + rp * ES_PAD + l);
  }
#endif
}

__device__ __forceinline__ void es_wait_tile() {
#if ES_TDM
  __builtin_amdgcn_s_wait_tensorcnt(0);   // y tile ready; stores (ASYNCcnt) keep flowing
#else
  es_wait_async0();                        // shared counter: drain everything
#endif
}

// 32 sequential ES steps for one row; circular seasonal buffer fully unrolled
// (constant indices -> SSA renaming, no v_mov shuffle cost).
template <bool FIRST>
__device__ __forceinline__ void es_compute_tile(
    const float* __restrict__ ylds,   // this thread's row in the y LDS tile
    float* __restrict__ lst,          // this thread's row in level staging
    float* __restrict__ sst,          // this thread's row in seasonality staging
    float& lev, float (&s)[ES_SEAS], float a, float b)
{
#pragma unroll
  for (int tt = 0; tt < ES_TT; ++tt) {
    float yv = ylds[tt];
    if (FIRST && tt == 0) {
      // t = 0: lev0 = y0 / init_full[0]; staged seas col 7 = init_full[7] = s[6]
      lev = yv / s[6];
      lst[0] = lev;
      sst[0] = s[6];
    } else {
      float season = s[0];
      float nl = a * (yv / season) + (1.0f - a) * lev;
      float ns = b * (yv / nl)     + (1.0f - b) * season;
      s[0] = s[1]; s[1] = s[2]; s[2] = s[3];
      s[3] = s[4]; s[4] = s[5]; s[5] = s[6]; s[6] = ns;
      lev = nl;
      lst[tt] = nl;
      sst[tt] = ns;
    }
  }
}

extern "C" __global__ __launch_bounds__(ES_RPB)
void es_scan_kernel(const float* __restrict__ y,
                    const float* __restrict__ lev_sms,
                    const float* __restrict__ seas_sms,
                    const float* __restrict__ init_seas,
                    const int*   __restrict__ idxs,
                    float* __restrict__ lev_out,
                    float* __restrict__ seas_out)
{
  extern __shared__ float sm[];
  float* yb   = sm;                     // 2 * ES_TILEF  (double-buffered y tiles)
  float* lstg = sm + 2 * ES_TILEF;      // 2 * ES_TILEF  (double-buffered level staging)
  float* sstg = sm + 4 * ES_TILEF;      // 2 * ES_TILEF  (double-buffered seas staging)

  const int tid  = threadIdx.x;
  const int w    = tid >> 5;            // wave id (0..1)
  const int l    = tid & 31;            // lane id
  const int row0 = blockIdx.x * ES_RPB;
  const int gr   = row0 + tid;          // this thread's global row

  // ---- kick off DMA of time-tile 0 ----
  es_issue_tile_load(y, row0, w, l, 0, yb);

  // ---- per-row init (overlaps with tile-0 DMA) ----
  const int idx = idxs[gr];
  const float a = 1.0f / (1.0f + __expf(-lev_sms[idx]));
  const float b = 1.0f / (1.0f + __expf(-seas_sms[idx]));

  float e[ES_SEAS];
#pragma unroll
  for (int j = 0; j < ES_SEAS; ++j) e[j] = __expf(init_seas[idx * ES_SEAS + j]);

  {
    // seasonalities[:, 0:8] = init_full = [e0..e6, e0]
    const size_t sb = (size_t)gr * ES_SLEN;
#pragma unroll
    for (int j = 0; j < ES_SEAS; ++j) seas_out[sb + j] = e[j];
    seas_out[sb + ES_SEAS] = e[0];
  }

  // buf0 = init_full[1:] = [e1..e6, e0]; lev0 uses init_full[0] = e0 = s[6]
  float s[ES_SEAS];
#pragma unroll
  for (int j = 0; j < ES_SEAS - 1; ++j) s[j] = e[j + 1];
  s[ES_SEAS - 1] = e[0];
  float lev = 0.0f;

  for (int k = 0; k < ES_NTILES; ++k) {
    // y tile k ready (issued at k-1). Staging reuse is safe WITHOUT an async
    // wait: ASYNCcnt drains stores in order and stalls issue at 63 outstanding,
    // so once all 64 stores of tile k-1 are issued (before this barrier),
    // every store of tile k-2 — the last readers of staging[k&1] — is complete.
    es_wait_tile();
    __syncthreads();

    // prefetch tile k+1 into the other y buffer
    if (k + 1 < ES_NTILES)
      es_issue_tile_load(y, row0, w, l, (k + 1) * ES_TT, yb + ((k + 1) & 1) * ES_TILEF);

    // sequential recurrence on registers, y from conflict-free padded LDS
    const float* ylds = yb + (k & 1) * ES_TILEF + tid * ES_PAD;
    float* myl = lstg + (k & 1) * ES_TILEF + tid * ES_PAD;
    float* mys = sstg + (k & 1) * ES_TILEF + tid * ES_PAD;
    if (k == 0) es_compute_tile<true >(ylds, myl, mys, lev, s, a, b);
    else        es_compute_tile<false>(ylds, myl, mys, lev, s, a, b);

    __syncthreads();  // staging visible to all waves

    // coalesced async stores straight out of LDS:
    //   levels col = t0 + l ; seasonalities col = t0 + 7 + l
    {
      const int t0 = k * ES_TT;
      float* lsrc = lstg + (k & 1) * ES_TILEF;
      float* ssrc = sstg + (k & 1) * ES_TILEF;
      float* lg = lev_out  + (size_t)row0 * ES_NTIME + t0 + l;
      float* sg = seas_out + (size_t)row0 * ES_SLEN  + t0 + ES_SEAS + l;
      for (int p = 0; p < ES_RPB / 2; ++p) {
        int rp = p * 2 + w;
        es_async_store(lg + (size_t)rp * ES_NTIME, lsrc + rp * ES_PAD + l);
        es_async_store(sg + (size_t)rp * ES_SLEN,  ssrc + rp * ES_PAD + l);
      }
    }
    // no trailing barrier: next loop-top barrier + issue-order backpressure cover reuse
  }
  // S_ENDPGM performs an implicit wait-idle, draining the final async stores.
}

extern "C" void kernel_launch(void* const* d_in, const int* in_sizes, int n_in,
                              void* d_out, int out_size, void* d_ws, size_t ws_size,
                              hipStream_t stream) {
  (void)in_sizes; (void)n_in; (void)out_size; (void)d_ws; (void)ws_size;

  const float* y        = (const float*)d_in[0];
  const float* lev_sms  = (const float*)d_in[1];
  const float* seas_sms = (const float*)d_in[2];
  const float* init_s   = (const float*)d_in[3];
  const int*   idxs     = (const int*)d_in[4];

  float* lev_out  = (float*)d_out;
  float* seas_out = lev_out + (size_t)ES_BATCH * ES_NTIME;

  const size_t smem_bytes = (size_t)(6 * ES_TILEF) * sizeof(float); // 50,688 B

  es_scan_kernel<<<ES_BATCH / ES_RPB, ES_RPB, smem_bytes, stream>>>(
      y, lev_sms, seas_sms, init_s, idxs, lev_out, seas_out);
}